// ColumnParallelLinearWithLoRA_7791070675058
// MI455X (gfx1250) — compile-verified
//
#include <hip/hip_runtime.h>

// Problem constants (match reference)
#define S_TOK  8192
#define D_K    4096   // D_IN
#define D_N    4096   // D_OUT
#define R_RANK 16

// Tiling
#define BM  128
#define BN  128
#define BK  16
#define LDK 20        // padded LDS row stride (floats) -> conflict-free b64 frag loads

typedef __attribute__((ext_vector_type(2))) float v2f;
typedef __attribute__((ext_vector_type(4))) float v4f;
typedef __attribute__((ext_vector_type(8))) float v8f;

// ---------------------------------------------------------------------------
// Kernel 1: xa[s, r] = sum_d x[s,d] * A[id_s, r, d]   (negligible FLOPs)
// One wave32 per token: lanes 0..15 -> r with K-half 0, lanes 16..31 -> r with
// K-half 1; combine halves with a cross-lane xor-16 shuffle.
// ---------------------------------------------------------------------------
__global__ __launch_bounds__(256) void lora_xa_kernel(
    const float* __restrict__ x, const int* __restrict__ ids,
    const float* __restrict__ A, float* __restrict__ xa)
{
    const int lane = threadIdx.x & 31;
    const int wid  = threadIdx.x >> 5;
    const int s    = blockIdx.x * 8 + wid;
    const int half = lane >> 4;
    const int r    = lane & 15;
    const int id   = ids[s];

    const v4f* xp = (const v4f*)(x + (size_t)s * D_K + half * (D_K / 2));
    const v4f* ap = (const v4f*)(A + ((size_t)id * R_RANK + r) * D_K + half * (D_K / 2));

    float acc = 0.f;
#pragma unroll 4
    for (int i = 0; i < D_K / 8; ++i) {   // 512 x float4 per half
        v4f xv = xp[i];
        v4f av = ap[i];
        acc += xv[0] * av[0] + xv[1] * av[1] + xv[2] * av[2] + xv[3] * av[3];
    }
    acc += __shfl_xor(acc, 16, 32);
    if (half == 0) xa[(size_t)s * R_RANK + r] = acc;
}

// ---------------------------------------------------------------------------
// Kernel 2: out = x @ W^T + bias + xa @ B[id]^T
// WMMA f32 16x16x4 main loop, double-buffered LDS, one barrier per k-tile,
// LoRA delta + bias fused into the epilogue.
// Block = 256 threads = 8 waves in a 4(M) x 2(N) grid; wave tile = 32x64.
// ---------------------------------------------------------------------------
__global__ __launch_bounds__(256) void lora_gemm_kernel(
    const float* __restrict__ x, const float* __restrict__ w,
    const float* __restrict__ bias, const float* __restrict__ Bbuf,
    const int* __restrict__ ids, const float* __restrict__ xa,
    float* __restrict__ out)
{
    __shared__ float sA[2][BM * LDK];   // x tile   [128 rows][BK] (padded), double buffered
    __shared__ float sB[2][BN * LDK];   // W tile   [128 cols][BK] (padded), double buffered
    __shared__ float sXA[BM * R_RANK];  // xa rows for this M tile
    __shared__ int   sID[BM];           // adapter ids for this M tile

    const int tid  = threadIdx.x;
    const int lane = tid & 31;
    const int wid  = tid >> 5;
    const int half = lane >> 4;   // K-half selector for WMMA frags
    const int l16  = lane & 15;
    const int wm   = wid >> 1;    // 0..3  (wave M position)
    const int wn   = wid & 1;     // 0..1  (wave N position)

    const int block_m = blockIdx.y * BM;
    const int block_n = blockIdx.x * BN;

    // Stage adapter ids + xa rows once per block (used only in epilogue).
    if (tid < BM) sID[tid] = ids[block_m + tid];
    {
        const int r  = tid >> 1;
        const int cg = (tid & 1) * 8;
        const float* src = xa + (size_t)(block_m + r) * R_RANK + cg;
        *(v4f*)&sXA[r * R_RANK + cg]     = *(const v4f*)(src);
        *(v4f*)&sXA[r * R_RANK + cg + 4] = *(const v4f*)(src + 4);
    }

    // Global->LDS staging assignment: thread t loads 8 floats of row t/2.
    const int   grow = tid >> 1;
    const int   gcol = (tid & 1) * 8;
    const float* xsrc = x + (size_t)(block_m + grow) * D_K + gcol;
    const float* wsrc = w + (size_t)(block_n + grow) * D_K + gcol;
    const int sOff = grow * LDK + gcol;

    v8f acc[2][4] = {};   // 2 M-subtiles x 4 N-subtiles of 16x16 f32

    // Prologue: first k-tile into registers.
    v4f ga0 = *(const v4f*)(xsrc);
    v4f ga1 = *(const v4f*)(xsrc + 4);
    v4f gb0 = *(const v4f*)(wsrc);
    v4f gb1 = *(const v4f*)(wsrc + 4);

    const int NKT = D_K / BK;   // 256 k-tiles
    for (int kt = 0; kt < NKT; ++kt) {
        const int buf = kt & 1;

        // Commit tile kt (held in registers since last iteration) to LDS.
        float* aDst = &sA[buf][sOff];
        float* bDst = &sB[buf][sOff];
        *(v4f*)(aDst)     = ga0;
        *(v4f*)(aDst + 4) = ga1;
        *(v4f*)(bDst)     = gb0;
        *(v4f*)(bDst + 4) = gb1;
        __syncthreads();   // single barrier: tile kt visible, tile kt-1 compute done

        // Issue global loads for tile kt+1 — consumed only at the NEXT store,
        // so the whole 32-WMMA compute phase hides their latency.
        if (kt + 1 < NKT) {
            const int k0n = (kt + 1) * BK;
            ga0 = *(const v4f*)(xsrc + k0n);
            ga1 = *(const v4f*)(xsrc + k0n + 4);
            gb0 = *(const v4f*)(wsrc + k0n);
            gb1 = *(const v4f*)(wsrc + k0n + 4);
            __builtin_prefetch(xsrc + k0n + BK, 0, 0);  // global_prefetch_b8
            __builtin_prefetch(wsrc + k0n + BK, 0, 0);
        }

        const float* sAb = sA[buf];
        const float* sBb = sB[buf];
        // 4 WMMA k-steps of K=4 each.
#pragma unroll
        for (int kk = 0; kk < 4; ++kk) {
            const int koff = kk * 4 + 2 * half;  // lanes 16-31 take K+2,K+3
            v2f afrag[2], bfrag[4];
#pragma unroll
            for (int im = 0; im < 2; ++im)
                afrag[im] = *(const v2f*)&sAb[(wm * 32 + im * 16 + l16) * LDK + koff];
#pragma unroll
            for (int in = 0; in < 4; ++in)
                bfrag[in] = *(const v2f*)&sBb[(wn * 64 + in * 16 + l16) * LDK + koff];
#pragma unroll
            for (int im = 0; im < 2; ++im)
#pragma unroll
                for (int in = 0; in < 4; ++in)
                    acc[im][in] = __builtin_amdgcn_wmma_f32_16x16x4_f32(
                        false, afrag[im], false, bfrag[in],
                        (short)0, acc[im][in], false, false);
        }
    }

    // Epilogue: out = acc + bias + sum_r xa[m,r] * B[id_m, n, r]
    float biasv[4];
#pragma unroll
    for (int in = 0; in < 4; ++in)
        biasv[in] = bias[block_n + wn * 64 + in * 16 + l16];

#pragma unroll
    for (int im = 0; im < 2; ++im) {
#pragma unroll
        for (int c = 0; c < 8; ++c) {
            // C/D layout: VGPR c -> M=c (lanes 0-15), M=c+8 (lanes 16-31)
            const int mloc = wm * 32 + im * 16 + half * 8 + c;
            const int mg   = block_m + mloc;
            const int id   = sID[mloc];
            v4f xa0 = *(const v4f*)&sXA[mloc * R_RANK];
            v4f xa1 = *(const v4f*)&sXA[mloc * R_RANK + 4];
            v4f xa2 = *(const v4f*)&sXA[mloc * R_RANK + 8];
            v4f xa3 = *(const v4f*)&sXA[mloc * R_RANK + 12];
#pragma unroll
            for (int in = 0; in < 4; ++in) {
                const int ng = block_n + wn * 64 + in * 16 + l16;
                const float* bp = Bbuf + ((size_t)id * D_N + ng) * R_RANK;
                v4f b0 = *(const v4f*)(bp);
                v4f b1 = *(const v4f*)(bp + 4);
                v4f b2 = *(const v4f*)(bp + 8);
                v4f b3 = *(const v4f*)(bp + 12);
                float delta = 0.f;
#pragma unroll
                for (int j = 0; j < 4; ++j)
                    delta += xa0[j] * b0[j] + xa1[j] * b1[j]
                           + xa2[j] * b2[j] + xa3[j] * b3[j];
                out[(size_t)mg * D_N + ng] = acc[im][in][c] + biasv[in] + delta;
            }
        }
    }
}

extern "C" void kernel_launch(void* const* d_in, const int* in_sizes, int n_in,
                              void* d_out, int out_size, void* d_ws, size_t ws_size,
                              hipStream_t stream) {
    const float* x    = (const float*)d_in[0];
    const int*   ids  = (const int*)d_in[1];
    const float* w    = (const float*)d_in[2];
    const float* bias = (const float*)d_in[3];
    const float* A    = (const float*)d_in[4];
    const float* B    = (const float*)d_in[5];
    float* out = (float*)d_out;
    float* xa  = (float*)d_ws;   // S_TOK * R_RANK floats = 512 KB scratch

    lora_xa_kernel<<<S_TOK / 8, 256, 0, stream>>>(x, ids, A, xa);

    dim3 grid(D_N / BN, S_TOK / BM);   // 32 x 64 blocks
    lora_gemm_kernel<<<grid, 256, 0, stream>>>(x, w, bias, B, ids, xa, out);
}